// Net_4174708212167
// MI455X (gfx1250) — compile-verified
//
#include <hip/hip_runtime.h>
#include <hip/hip_bf16.h>

// ---------------------------------------------------------------------------
// Quantum-patch feature extractor + MLP for MI455X (gfx1250, wave32).
//   Kernel 1: one thread per 2x2 patch, 4-qubit (16 complex amp) simulation
//             fully unrolled in VGPRs -> feat[128][784] (fp32, in d_ws)
//   Kernel 2: fc1  (128x784)@(784x64)^T + bias, ReLU  via v_wmma_f32_16x16x4_f32
//   Kernel 3: fc2  (128x64)@(64x10)^T + bias (N padded to 16) via same WMMA
// ---------------------------------------------------------------------------

typedef __attribute__((ext_vector_type(2))) float v2f;
typedef __attribute__((ext_vector_type(8))) float v8f;

#define TWO_PI 6.283185307179586f

// ---------------- circuit gate helpers (Q compile-time => reg-resident state)
template <int Q>
__device__ __forceinline__ void apply_rx(float* re, float* im, float c, float s) {
  constexpr int st = 8 >> Q;
#pragma unroll
  for (int s0 = 0; s0 < 16; ++s0) {
    if (s0 & st) continue;
    const int s1 = s0 | st;
    float a0r = re[s0], a0i = im[s0], a1r = re[s1], a1i = im[s1];
    // Rx: [[c, -i s],[-i s, c]]
    re[s0] = c * a0r + s * a1i;  im[s0] = c * a0i - s * a1r;
    re[s1] = c * a1r + s * a0i;  im[s1] = c * a1i - s * a0r;
  }
}

template <int Q>
__device__ __forceinline__ void apply_ry(float* re, float* im, float c, float s) {
  constexpr int st = 8 >> Q;
#pragma unroll
  for (int s0 = 0; s0 < 16; ++s0) {
    if (s0 & st) continue;
    const int s1 = s0 | st;
    float a0r = re[s0], a0i = im[s0], a1r = re[s1], a1i = im[s1];
    // Ry: [[c,-s],[s,c]] (real)
    re[s0] = c * a0r - s * a1r;  im[s0] = c * a0i - s * a1i;
    re[s1] = s * a0r + c * a1r;  im[s1] = s * a0i + c * a1i;
  }
}

template <int Q>
__device__ __forceinline__ void apply_rz(float* re, float* im, float c, float s) {
  constexpr int st = 8 >> Q;
#pragma unroll
  for (int s0 = 0; s0 < 16; ++s0) {
    if (s0 & st) continue;
    const int s1 = s0 | st;
    float a0r = re[s0], a0i = im[s0], a1r = re[s1], a1i = im[s1];
    // Rz: diag(e^{-i t/2}, e^{+i t/2}) = diag((c,-s),(c,+s))
    re[s0] = c * a0r + s * a0i;  im[s0] = c * a0i - s * a0r;
    re[s1] = c * a1r - s * a1i;  im[s1] = c * a1i + s * a1r;
  }
}

template <int C, int T>
__device__ __forceinline__ void apply_cnot(float* re, float* im) {
  constexpr int mc = 8 >> C;
  constexpr int mt = 8 >> T;
#pragma unroll
  for (int s0 = 0; s0 < 16; ++s0) {
    if ((s0 & mc) && !(s0 & mt)) {
      const int s1 = s0 | mt;
      float tr = re[s0], ti = im[s0];
      re[s0] = re[s1]; im[s0] = im[s1];
      re[s1] = tr;     im[s1] = ti;
    }
  }
}

// ---------------- kernel 1: per-patch 4-qubit circuit -> features -----------
__global__ void circuit_kernel(const float* __restrict__ x,
                               const float* __restrict__ w,
                               float* __restrict__ feat) {
  const int p = blockIdx.x * blockDim.x + threadIdx.x;
  if (p >= 128 * 196) return;
  const int b = p / 196;
  const int r = p % 196;
  const int i = r / 14;
  const int j = r % 14;
  const float* img = x + b * 784;  // [28][28]

  float ang0 = TWO_PI * img[(2 * i) * 28 + 2 * j];
  float ang1 = TWO_PI * img[(2 * i) * 28 + 2 * j + 1];
  float ang2 = TWO_PI * img[(2 * i + 1) * 28 + 2 * j];
  float ang3 = TWO_PI * img[(2 * i + 1) * 28 + 2 * j + 1];

  float re[16], im[16];
#pragma unroll
  for (int s = 0; s < 16; ++s) { re[s] = 0.f; im[s] = 0.f; }
  re[0] = 1.f;

  // Rx encoding
  apply_rx<0>(re, im, __cosf(0.5f * ang0), __sinf(0.5f * ang0));
  apply_rx<1>(re, im, __cosf(0.5f * ang1), __sinf(0.5f * ang1));
  apply_rx<2>(re, im, __cosf(0.5f * ang2), __sinf(0.5f * ang2));
  apply_rx<3>(re, im, __cosf(0.5f * ang3), __sinf(0.5f * ang3));

  // 5 variational layers (Ry, Rz, Ry), ring CNOT between layers
#pragma unroll
  for (int layer = 0; layer < 5; ++layer) {
    const float* pw = w + layer * 12;
    apply_ry<0>(re, im, __cosf(0.5f * pw[0]), __sinf(0.5f * pw[0]));
    apply_ry<1>(re, im, __cosf(0.5f * pw[1]), __sinf(0.5f * pw[1]));
    apply_ry<2>(re, im, __cosf(0.5f * pw[2]), __sinf(0.5f * pw[2]));
    apply_ry<3>(re, im, __cosf(0.5f * pw[3]), __sinf(0.5f * pw[3]));
    apply_rz<0>(re, im, __cosf(0.5f * pw[4]), __sinf(0.5f * pw[4]));
    apply_rz<1>(re, im, __cosf(0.5f * pw[5]), __sinf(0.5f * pw[5]));
    apply_rz<2>(re, im, __cosf(0.5f * pw[6]), __sinf(0.5f * pw[6]));
    apply_rz<3>(re, im, __cosf(0.5f * pw[7]), __sinf(0.5f * pw[7]));
    apply_ry<0>(re, im, __cosf(0.5f * pw[8]), __sinf(0.5f * pw[8]));
    apply_ry<1>(re, im, __cosf(0.5f * pw[9]), __sinf(0.5f * pw[9]));
    apply_ry<2>(re, im, __cosf(0.5f * pw[10]), __sinf(0.5f * pw[10]));
    apply_ry<3>(re, im, __cosf(0.5f * pw[11]), __sinf(0.5f * pw[11]));
    if (layer < 4) {
      apply_cnot<0, 1>(re, im);
      apply_cnot<1, 2>(re, im);
      apply_cnot<2, 3>(re, im);
      apply_cnot<3, 0>(re, im);
    }
  }

  // <Z_q> expectations; feature layout matches reference reshape(B, 14*14*4)
  float* fout = feat + b * 784 + (i * 14 + j) * 4;
#pragma unroll
  for (int q = 0; q < 4; ++q) {
    const int bit = 8 >> q;
    float e = 0.f;
#pragma unroll
    for (int s = 0; s < 16; ++s) {
      float pr = re[s] * re[s] + im[s] * im[s];
      e += (s & bit) ? -pr : pr;
    }
    fout[q] = e;
  }
}

// ---------------- kernel 2: fc1 via full-precision f32 WMMA -----------------
// grid = 8 blocks (M tiles of 16), block = 128 threads (4 waves = 4 N tiles).
// A 16x4 fp32 layout: lane<16 -> A[M=lane][K=0],A[...][K=1] in the two VGPRs;
//                     lane>=16 -> K=2,3.  B (4x16) mirrored by K-half.
// D 16x16 fp32: VGPR r -> row M=r (lanes 0-15) / M=r+8 (lanes 16-31).
__global__ void fc1_wmma(const float* __restrict__ feat,
                         const float* __restrict__ w,     // [64][784]
                         const float* __restrict__ bias,  // [64]
                         float* __restrict__ h) {         // [128][64]
  const int wave = threadIdx.x >> 5;
  const int lane = threadIdx.x & 31;
  const int m0 = blockIdx.x * 16;
  const int n0 = wave * 16;
  const int lrow = lane & 15;
  const int khalf = (lane >> 4) << 1;  // 0 or 2

  const float* arow = feat + (m0 + lrow) * 784 + khalf;
  const float* brow = w + (n0 + lrow) * 784 + khalf;

  v8f acc = {};
  for (int k0 = 0; k0 < 784; k0 += 4) {
    v2f a, b;
    a.x = arow[k0];
    a.y = arow[k0 + 1];
    b.x = brow[k0];
    b.y = brow[k0 + 1];
    acc = __builtin_amdgcn_wmma_f32_16x16x4_f32(
        /*neg_a=*/false, a, /*neg_b=*/false, b,
        /*c_mod=*/(short)0, acc, /*reuse_a=*/false, /*reuse_b=*/false);
  }

  const int nOut = n0 + lrow;
  const int mBase = m0 + ((lane >> 4) << 3);
  const float bv = bias[nOut];
#pragma unroll
  for (int rr = 0; rr < 8; ++rr) {
    float v = acc[rr] + bv;
    h[(mBase + rr) * 64 + nOut] = fmaxf(v, 0.f);
  }
}

// ---------------- kernel 3: fc2 via f32 WMMA, N padded 10 -> 16 -------------
// grid = 8 blocks (M tiles), block = 32 threads (one wave).
__global__ void fc2_wmma(const float* __restrict__ h,     // [128][64]
                         const float* __restrict__ w,     // [10][64]
                         const float* __restrict__ bias,  // [10]
                         float* __restrict__ out) {       // [128][10]
  const int lane = threadIdx.x & 31;
  const int m0 = blockIdx.x * 16;
  const int lrow = lane & 15;
  const int khalf = (lane >> 4) << 1;
  const bool nvalid = (lrow < 10);

  const float* arow = h + (m0 + lrow) * 64 + khalf;
  const float* brow = w + lrow * 64 + khalf;  // row lrow of fc2_w

  v8f acc = {};
  for (int k0 = 0; k0 < 64; k0 += 4) {
    v2f a, b;
    a.x = arow[k0];
    a.y = arow[k0 + 1];
    b.x = nvalid ? brow[k0] : 0.f;      // pad N=10..15 with zero columns
    b.y = nvalid ? brow[k0 + 1] : 0.f;
    acc = __builtin_amdgcn_wmma_f32_16x16x4_f32(
        false, a, false, b, (short)0, acc, false, false);
  }

  if (nvalid) {
    const float bv = bias[lrow];
    const int mBase = m0 + ((lane >> 4) << 3);
#pragma unroll
    for (int rr = 0; rr < 8; ++rr) {
      out[(mBase + rr) * 10 + lrow] = acc[rr] + bv;
    }
  }
}

// ---------------------------------------------------------------------------
extern "C" void kernel_launch(void* const* d_in, const int* in_sizes, int n_in,
                              void* d_out, int out_size, void* d_ws, size_t ws_size,
                              hipStream_t stream) {
  const float* x      = (const float*)d_in[0];  // [128,1,28,28]
  const float* weight = (const float*)d_in[1];  // [60]
  const float* fc1_w  = (const float*)d_in[2];  // [64,784]
  const float* fc1_b  = (const float*)d_in[3];  // [64]
  const float* fc2_w  = (const float*)d_in[4];  // [10,64]
  const float* fc2_b  = (const float*)d_in[5];  // [10]
  float* out = (float*)d_out;                   // [128,10]

  float* feat = (float*)d_ws;                   // [128][784]  (401 KB)
  float* h    = feat + 128 * 784;               // [128][64]   ( 32 KB)

  const int npatch = 128 * 196;
  circuit_kernel<<<(npatch + 255) / 256, 256, 0, stream>>>(x, weight, feat);
  fc1_wmma<<<8, 128, 0, stream>>>(feat, fc1_w, fc1_b, h);
  fc2_wmma<<<8, 32, 0, stream>>>(h, fc2_w, fc2_b, out);
}